// GATLayer_36876589204178
// MI455X (gfx1250) — compile-verified
//
#include <hip/hip_runtime.h>
#include <math.h>
#include <stdint.h>

#define BB   8
#define NN   2048
#define FIN  128
#define FOUT 64
#define ALPHA 0.2f
#define NEG_INF (-9.0e15f)
#define LN_EPS 1e-5f

typedef __attribute__((ext_vector_type(2))) float v2f;
typedef __attribute__((ext_vector_type(8))) float v8f;

// ---------------------------------------------------------------------------
// Kernel 1: h = x @ W  (B*N x FIN) @ (FIN x FOUT) via fp32 WMMA 16x16x4.
// Grid: (B*N)/16 blocks, 128 threads (4 waves). Wave w computes the 16x16
// output tile at columns [16w, 16w+16).
// ---------------------------------------------------------------------------
__global__ void gemm_h(const float* __restrict__ x, const float* __restrict__ W,
                       float* __restrict__ h) {
    const int wave = threadIdx.x >> 5;
    const int lane = threadIdx.x & 31;
    const int lo = lane & 15;        // M (A) / N (B,C)
    const int hi = lane >> 4;        // half-wave select
    const int row0 = blockIdx.x * 16;           // global row in [0, B*N)
    const int n0 = wave * 16;

    const float* xrow = x + (size_t)(row0 + lo) * FIN;

    v8f c = {0.f,0.f,0.f,0.f,0.f,0.f,0.f,0.f};
    #pragma unroll 4
    for (int kk = 0; kk < FIN; kk += 4) {
        const int ka = kk + hi * 2;
        // A fragment: lane holds A[M=lo][K=ka], A[M=lo][K=ka+1] (contiguous pair)
        v2f a = *(const v2f*)(xrow + ka);
        // B fragment: lane holds B[K=ka][N=n0+lo], B[K=ka+1][N=n0+lo]
        v2f b;
        b.x = W[(size_t)ka * FOUT + n0 + lo];
        b.y = W[(size_t)(ka + 1) * FOUT + n0 + lo];
        c = __builtin_amdgcn_wmma_f32_16x16x4_f32(false, a, false, b,
                                                  (short)0, c, false, false);
    }
    // C layout: VGPR r <-> M = r + 8*hi, N = lo
    float* hout = h + (size_t)row0 * FOUT + n0;
    #pragma unroll
    for (int r = 0; r < 8; ++r) {
        hout[(size_t)(r + 8 * hi) * FOUT + lo] = c[r];
    }
}

// ---------------------------------------------------------------------------
// Kernel 2: s1 = h @ a1, s2 = h @ a2 (one thread per node row).
// ---------------------------------------------------------------------------
__global__ void score_kernel(const float* __restrict__ h, const float* __restrict__ a,
                             float* __restrict__ s1, float* __restrict__ s2) {
    const int row = blockIdx.x * blockDim.x + threadIdx.x;
    if (row >= BB * NN) return;
    const float* hr = h + (size_t)row * FOUT;
    float acc1 = 0.f, acc2 = 0.f;
    #pragma unroll 8
    for (int f = 0; f < FOUT; ++f) {
        const float v = hr[f];
        acc1 += v * a[f];
        acc2 += v * a[FOUT + f];
    }
    s1[row] = acc1;
    s2[row] = acc2;
}

// ---------------------------------------------------------------------------
// Kernel 3: fused masked-softmax attention + aggregation + LayerNorm + ELU.
// Grid: (B*N)/16 blocks (one 16-query tile each), 128 threads = 4 waves.
// Wave w processes 64-wide key blocks w, w+4, ..., double-buffering the adj
// tile (16x64 ints) into LDS with global_load_async_to_lds_b128 (fully
// coalesced 256B row segments, ASYNCcnt-tracked), then runs an online
// softmax and accumulates P(16x16) @ H(16x64) via fp32 WMMA.
// ---------------------------------------------------------------------------
__global__ void __launch_bounds__(128)
gat_attn(const float* __restrict__ h, const int* __restrict__ adj,
         const float* __restrict__ s1, const float* __restrict__ s2,
         const float* __restrict__ gamma, const float* __restrict__ beta,
         float* __restrict__ out) {
    __shared__ int   sm_adj[4][2][16][64];   // per-wave double-buffered adj tile
    __shared__ float sm_m[4][16];
    __shared__ float sm_l[4][16];
    __shared__ float sm_acc[4][16][FOUT];
    __shared__ float sm_hp[16][FOUT];

    const int wave = threadIdx.x >> 5;
    const int lane = threadIdx.x & 31;
    const int lo = lane & 15;
    const int hi = lane >> 4;

    const int b  = blockIdx.x / (NN / 16);
    const int i0 = (blockIdx.x % (NN / 16)) * 16;
    const int i  = i0 + lo;                    // this lane's query row (A layout)

    const float s1i = s1[(size_t)b * NN + i];
    const float* s2b = s2 + (size_t)b * NN;
    const float* hb  = h + (size_t)b * NN * FOUT;

    // Async-copy one 16x64 adj block (rows i0..i0+15, cols [t*64, t*64+64))
    // into sm_adj[wave][buf]. Instruction q: lanes 0-15 -> row 2q (256B
    // contiguous), lanes 16-31 -> row 2q+1 (256B contiguous).
    auto issue_async = [&](int tblk, int buf) {
        uint64_t g = (uint64_t)adj
                   + (((uint64_t)(b * NN + i0 + hi) * NN + (uint64_t)tblk * 64) << 2)
                   + (unsigned)(lo * 16);
        unsigned ldsa = (unsigned)(uintptr_t)(&sm_adj[wave][buf][0][0])
                      + (unsigned)(hi * 256 + lo * 16);
        #pragma unroll
        for (int q = 0; q < 8; ++q) {
            asm volatile("global_load_async_to_lds_b128 %0, %1, off"
                         :: "v"(ldsa), "v"(g)
                         : "memory");
            g += (uint64_t)2 * NN * 4;   // advance 2 rows
            ldsa += 512;                 // 2 rows * 256B in LDS
        }
    };

    float m = -INFINITY;
    float l = 0.f;
    v8f acc[4];
    #pragma unroll
    for (int nt = 0; nt < 4; ++nt) acc[nt] = (v8f){0.f,0.f,0.f,0.f,0.f,0.f,0.f,0.f};

    const int NKB = (NN / 64) / 4;   // 8 blocks per wave
    issue_async(wave, 0);

    for (int kb = 0; kb < NKB; ++kb) {
        const int t = wave + 4 * kb;
        if (kb + 1 < NKB) {
            issue_async(wave + 4 * (kb + 1), (kb + 1) & 1);
            asm volatile("s_wait_asynccnt 0x8" ::: "memory");  // current buffer landed
        } else {
            asm volatile("s_wait_asynccnt 0x0" ::: "memory");
        }
        const int (*smadj)[64] = sm_adj[wave][kb & 1];
        const int j0b = t * 64;

        #pragma unroll
        for (int st = 0; st < 4; ++st) {
            // ---- scores for this 16-wide sub-tile, WMMA A-fragment order ----
            float att8[8];
            float mx = -INFINITY;
            #pragma unroll
            for (int kk = 0; kk < 4; ++kk) {
                const int jc = 16 * st + 4 * kk + 2 * hi;   // col within block
                const int j  = j0b + jc;
                v2f sj = *(const v2f*)(s2b + j);
                int2 av = *(const int2*)&smadj[lo][jc];     // ds_load_b64
                float e0 = s1i + sj.x; e0 = e0 > 0.f ? e0 : ALPHA * e0;
                float e1 = s1i + sj.y; e1 = e1 > 0.f ? e1 : ALPHA * e1;
                att8[2 * kk]     = av.x > 0 ? e0 : NEG_INF;
                att8[2 * kk + 1] = av.y > 0 ? e1 : NEG_INF;
                mx = fmaxf(mx, fmaxf(att8[2 * kk], att8[2 * kk + 1]));
            }
            // full row max across the two half-wave lanes holding this row
            mx = fmaxf(mx, __shfl_xor(mx, 16));
            const float m_new = fmaxf(m, mx);
            const float rowscale = __expf(m - m_new);   // exp(-inf - finite)=0 first time

            // ---- probabilities + running denominator -----------------------
            float p8[8];
            float psum = 0.f;
            #pragma unroll
            for (int q = 0; q < 8; ++q) {
                const float p = __expf(att8[q] - m_new);
                p8[q] = p;
                psum += p;
            }
            psum += __shfl_xor(psum, 16);
            l = l * rowscale + psum;
            m = m_new;

            // ---- rescale accumulator (C layout rows r + 8*hi) --------------
            float rs[8];
            #pragma unroll
            for (int r = 0; r < 8; ++r) rs[r] = __shfl(rowscale, r + 8 * hi);
            #pragma unroll
            for (int nt = 0; nt < 4; ++nt) {
                #pragma unroll
                for (int r = 0; r < 8; ++r) acc[nt][r] *= rs[r];
            }

            // ---- acc += P(16x16) @ H(16x64) via 16 chained WMMAs -----------
            #pragma unroll
            for (int nt = 0; nt < 4; ++nt) {
                const int n0 = nt * 16;
                #pragma unroll
                for (int kk = 0; kk < 4; ++kk) {
                    const int j = j0b + 16 * st + 4 * kk + 2 * hi;
                    v2f bf;
                    bf.x = hb[(size_t)j * FOUT + n0 + lo];
                    bf.y = hb[(size_t)(j + 1) * FOUT + n0 + lo];
                    v2f af;
                    af.x = p8[2 * kk];
                    af.y = p8[2 * kk + 1];
                    acc[nt] = __builtin_amdgcn_wmma_f32_16x16x4_f32(
                        false, af, false, bf, (short)0, acc[nt], false, false);
                }
            }
        }
    }

    // ---- spill per-wave state to LDS ---------------------------------------
    if (lane < 16) {
        sm_m[wave][lo] = m;
        sm_l[wave][lo] = l;
    }
    #pragma unroll
    for (int nt = 0; nt < 4; ++nt) {
        #pragma unroll
        for (int r = 0; r < 8; ++r) {
            sm_acc[wave][r + 8 * hi][nt * 16 + lo] = acc[nt][r];
        }
    }
    __syncthreads();

    // ---- merge 4 waves: thread t handles row t/8, feature block (t%8)*8 ----
    {
        const int row = threadIdx.x >> 3;
        const int fb = (threadIdx.x & 7) * 8;
        float M = fmaxf(fmaxf(sm_m[0][row], sm_m[1][row]),
                        fmaxf(sm_m[2][row], sm_m[3][row]));
        float wsc[4];
        float L = 0.f;
        #pragma unroll
        for (int w = 0; w < 4; ++w) {
            wsc[w] = __expf(sm_m[w][row] - M);
            L += sm_l[w][row] * wsc[w];
        }
        const float invL = 1.f / L;
        #pragma unroll
        for (int f = fb; f < fb + 8; ++f) {
            float v = 0.f;
            #pragma unroll
            for (int w = 0; w < 4; ++w) v += sm_acc[w][row][f] * wsc[w];
            sm_hp[row][f] = v * invL;
        }
    }
    __syncthreads();

    // ---- LayerNorm over FOUT + ELU, one row per thread (t < 16) ------------
    if (threadIdx.x < 16) {
        const int r = threadIdx.x;
        float mu = 0.f;
        #pragma unroll 8
        for (int f = 0; f < FOUT; ++f) mu += sm_hp[r][f];
        mu *= (1.f / FOUT);
        float var = 0.f;
        #pragma unroll 8
        for (int f = 0; f < FOUT; ++f) {
            const float d = sm_hp[r][f] - mu;
            var += d * d;
        }
        var *= (1.f / FOUT);
        const float inv = rsqrtf(var + LN_EPS);
        float* outp = out + ((size_t)b * NN + i0 + r) * FOUT;
        #pragma unroll 8
        for (int f = 0; f < FOUT; ++f) {
            float y = (sm_hp[r][f] - mu) * inv * gamma[f] + beta[f];
            outp[f] = y > 0.f ? y : (__expf(y) - 1.f);
        }
    }
}

// ---------------------------------------------------------------------------
extern "C" void kernel_launch(void* const* d_in, const int* in_sizes, int n_in,
                              void* d_out, int out_size, void* d_ws, size_t ws_size,
                              hipStream_t stream) {
    const float* x     = (const float*)d_in[0];
    const int*   adj   = (const int*)d_in[1];
    const float* W     = (const float*)d_in[2];
    const float* a     = (const float*)d_in[3];
    const float* gamma = (const float*)d_in[4];
    const float* beta  = (const float*)d_in[5];
    float* out = (float*)d_out;

    float* h  = (float*)d_ws;                       // B*N*FOUT = 4 MB
    float* s1 = h + (size_t)BB * NN * FOUT;         // B*N
    float* s2 = s1 + (size_t)BB * NN;               // B*N

    gemm_h<<<(BB * NN) / 16, 128, 0, stream>>>(x, W, h);
    score_kernel<<<(BB * NN + 255) / 256, 256, 0, stream>>>(h, a, s1, s2);
    gat_attn<<<(BB * NN) / 16, 128, 0, stream>>>(h, adj, s1, s2, gamma, beta, out);
}